// multiTimeAttention_240518168902
// MI455X (gfx1250) — compile-verified
//
#include <hip/hip_runtime.h>
#include <hip/hip_bf16.h>
#include <math.h>

// ---------------------------------------------------------------------------
// multiTimeAttention for MI455X (gfx1250, wave32, WMMA)
//
// Pipeline:
//   1) qp = query @ Wq + bq            (WMMA f32 16x16x4)
//   2) kp = key_ts @ Wk + bk           (WMMA f32 16x16x4)
//   3) scores[b,h] = qp_h @ kp_h^T / sqrt(32)   (WMMA f32 16x16x4)
//   4) online masked softmax over Lk per (b,q,h,d) + weighted reduce -> x
//   5) W3 row L1 norms -> stable descending rank -> gather W3p
//   6) out = x @ W3p + b3              (WMMA f32 16x16x4)
// ---------------------------------------------------------------------------

typedef __attribute__((ext_vector_type(2))) float v2f;
typedef __attribute__((ext_vector_type(8))) float v8f;

#define EMBED_TIME 128
#define NUM_HEADS  4
#define DK         32        // EMBED_TIME / NUM_HEADS
#define INPUT_DIM  64
#define NHIDDEN    256
#define BB         4
#define LQ         256
#define LK         512
#define HD         256       // NUM_HEADS * INPUT_DIM

// ---------------------------------------------------------------------------
// Generic GEMM: C[M,N] = alpha * (A[M,K] @ B[K,N]) + bias[N]
// A row-major lda=K, B row-major ldb=N. One wave per 16x16 C tile.
// Requires M%16==0, N%16==0, K%4==0. EXEC is all-1s (full 32-lane blocks).
// ---------------------------------------------------------------------------
__global__ __launch_bounds__(32)
void gemm_wmma_f32(const float* __restrict__ A, const float* __restrict__ B,
                   const float* __restrict__ bias, float* __restrict__ C,
                   int M, int N, int K, float alpha)
{
    const int ntn  = N >> 4;
    const int tile = blockIdx.x;
    const int tm   = (tile / ntn) << 4;
    const int tn   = (tile % ntn) << 4;
    const int lane = threadIdx.x;        // 0..31
    const int mr   = lane & 15;          // m (for A) / n (for B,C)
    const int kk0  = (lane >> 4) << 1;   // 0 or 2: K sub-offset this half-wave

    v8f acc = {};
    const float* arow = A + (size_t)(tm + mr) * K;       // A row for this lane
    const float* bcol = B + (size_t)(tn + mr);           // B column for this lane
    for (int k = 0; k < K; k += 4) {
        v2f a, b;
        a.x = arow[k + kk0];
        a.y = arow[k + kk0 + 1];
        b.x = bcol[(size_t)(k + kk0) * N];
        b.y = bcol[(size_t)(k + kk0 + 1) * N];
        acc = __builtin_amdgcn_wmma_f32_16x16x4_f32(
            /*neg_a=*/false, a, /*neg_b=*/false, b,
            /*c_mod=*/(short)0, acc, /*reuse_a=*/false, /*reuse_b=*/false);
    }

    const int   cn    = tn + mr;
    const float bv    = bias ? bias[cn] : 0.0f;
    const int   mbase = tm + ((lane >> 4) << 3);   // rows 0..7 or 8..15 of tile
#pragma unroll
    for (int r = 0; r < 8; ++r)
        C[(size_t)(mbase + r) * N + cn] = acc[r] * alpha + bv;
}

// ---------------------------------------------------------------------------
// Batched scores: S[b,h,q,k] = scale * sum_e qp[b,q,h*32+e] * kp[b,k,h*32+e]
// A = qp (row-major, lda=128), effective B = kp^T so both fragments are
// contiguous float2 loads. grid.x = (LQ/16)*(LK/16) = 512 tiles, grid.y = B*H.
// ---------------------------------------------------------------------------
__global__ __launch_bounds__(32)
void scores_wmma_f32(const float* __restrict__ QP, const float* __restrict__ KP,
                     float* __restrict__ S, float scale)
{
    const int bh   = blockIdx.y;
    const int b    = bh >> 2;
    const int h    = bh & 3;
    const int tile = blockIdx.x;
    const int tn   = (tile & 31) << 4;   // 32 tiles along Lk
    const int tm   = (tile >> 5) << 4;   // 16 tiles along Lq
    const int lane = threadIdx.x;
    const int mr   = lane & 15;
    const int kk0  = (lane >> 4) << 1;

    const float* arow = QP + ((size_t)b * LQ + tm + mr) * EMBED_TIME + h * DK;
    const float* brow = KP + ((size_t)b * LK + tn + mr) * EMBED_TIME + h * DK;

    v8f acc = {};
#pragma unroll
    for (int k = 0; k < DK; k += 4) {
        v2f a, bfr;
        a.x   = arow[k + kk0];
        a.y   = arow[k + kk0 + 1];
        bfr.x = brow[k + kk0];        // B[e][n] = KP[b][tn+n][h*32+e]
        bfr.y = brow[k + kk0 + 1];
        acc = __builtin_amdgcn_wmma_f32_16x16x4_f32(
            false, a, false, bfr, (short)0, acc, false, false);
    }

    float* crow = S + (size_t)bh * LQ * LK;
    const int cn    = tn + mr;
    const int mbase = tm + ((lane >> 4) << 3);
#pragma unroll
    for (int r = 0; r < 8; ++r)
        crow[(size_t)(mbase + r) * LK + cn] = acc[r] * scale;
}

// ---------------------------------------------------------------------------
// Online masked softmax over Lk + weighted reduce with value.
// One block per (b,q); thread t -> (h = t/64, d = t%64).
// Uses the reference's exact -1e9 masking so the all-masked edge case
// degenerates to a uniform average, identical to JAX.
// ---------------------------------------------------------------------------
__global__ __launch_bounds__(256)
void mta_softmax_reduce(const float* __restrict__ S, const float* __restrict__ V,
                        const int* __restrict__ MK, float* __restrict__ X)
{
    const int b = blockIdx.x >> 8;   // / LQ
    const int q = blockIdx.x & 255;  // % LQ

    __shared__ float s_sh[NUM_HEADS * LK];   // 8 KB
    for (int i = threadIdx.x; i < NUM_HEADS * LK; i += 256) {
        const int h = i >> 9, k = i & (LK - 1);
        s_sh[i] = S[(((size_t)(b * NUM_HEADS + h) * LQ + q) << 9) + k];
    }
    __syncthreads();

    const int h = threadIdx.x >> 6;
    const int d = threadIdx.x & 63;

    const float* srow = s_sh + (h << 9);
    const float* vcol = V  + (size_t)b * LK * INPUT_DIM + d;
    const int*   mcol = MK + (size_t)b * LK * INPUT_DIM + d;

    float mmax = -3.0e38f;
    float num = 0.0f, den = 0.0f;
    for (int k = 0; k < LK; ++k) {
        const float s = mcol[(size_t)k * INPUT_DIM] ? srow[k] : -1.0e9f;
        const float v = vcol[(size_t)k * INPUT_DIM];
        const float mnew  = fmaxf(mmax, s);
        const float scl   = __expf(mmax - mnew);
        const float p     = __expf(s - mnew);
        num  = num * scl + p * v;
        den  = den * scl + p;
        mmax = mnew;
    }
    X[((size_t)b * LQ + q) * HD + (h << 6) + d] = num / den;
}

// ---------------------------------------------------------------------------
// W3 row L1 norms (one block per row, tree reduction).
// ---------------------------------------------------------------------------
__global__ __launch_bounds__(256)
void w3_rownorm(const float* __restrict__ W3, float* __restrict__ nrm)
{
    __shared__ float red[256];
    const int i = blockIdx.x;
    red[threadIdx.x] = fabsf(W3[(size_t)i * NHIDDEN + threadIdx.x]);
    __syncthreads();
    for (int s = 128; s > 0; s >>= 1) {
        if (threadIdx.x < (unsigned)s) red[threadIdx.x] += red[threadIdx.x + s];
        __syncthreads();
    }
    if (threadIdx.x == 0) nrm[i] = red[0];
}

// ---------------------------------------------------------------------------
// Stable descending rank of row i, then scatter row i to W3p[rank].
// rank(i) = #{j : n_j > n_i} + #{j < i : n_j == n_i}  (stable argsort(-n)).
// ---------------------------------------------------------------------------
__global__ __launch_bounds__(256)
void w3_permute(const float* __restrict__ W3, const float* __restrict__ nrm,
                float* __restrict__ W3p)
{
    __shared__ int cnt[256];
    const int i  = blockIdx.x;
    const int j  = threadIdx.x;
    const float ni = nrm[i];
    const float nj = nrm[j];
    cnt[j] = (nj > ni || (nj == ni && j < i)) ? 1 : 0;
    __syncthreads();
    for (int s = 128; s > 0; s >>= 1) {
        if (j < s) cnt[j] += cnt[j + s];
        __syncthreads();
    }
    const int rank = cnt[0];
    W3p[(size_t)rank * NHIDDEN + j] = W3[(size_t)i * NHIDDEN + j];
}

// ---------------------------------------------------------------------------
// Orchestration
// ---------------------------------------------------------------------------
extern "C" void kernel_launch(void* const* d_in, const int* in_sizes, int n_in,
                              void* d_out, int out_size, void* d_ws, size_t ws_size,
                              hipStream_t stream)
{
    const float* query  = (const float*)d_in[0];   // [4,256,128]
    const float* key_ts = (const float*)d_in[1];   // [4,512,128]
    const float* value  = (const float*)d_in[2];   // [4,512,64]
    const int*   mask   = (const int*)  d_in[3];   // [4,512,64]
    const float* Wq     = (const float*)d_in[4];   // [128,128]
    const float* bq     = (const float*)d_in[5];
    const float* Wk     = (const float*)d_in[6];
    const float* bk     = (const float*)d_in[7];
    const float* W3     = (const float*)d_in[8];   // [256,256]
    const float* b3     = (const float*)d_in[9];
    float*       out    = (float*)d_out;           // [4,256,256]

    // Workspace layout (floats); total ~11 MB.
    float* ws     = (float*)d_ws;
    float* qp     = ws;                                  // 4*256*128   = 131072
    float* kp     = qp + (size_t)BB * LQ * EMBED_TIME;   // 4*512*128   = 262144
    float* scores = kp + (size_t)BB * LK * EMBED_TIME;   // 16*256*512  = 2097152
    float* x      = scores + (size_t)BB * NUM_HEADS * LQ * LK; // 1024*256
    float* W3p    = x + (size_t)BB * LQ * HD;            // 256*256
    float* nrm    = W3p + (size_t)HD * NHIDDEN;          // 256

    // 1) Q projection: [1024,128] @ [128,128] + bq
    gemm_wmma_f32<<<dim3((BB * LQ / 16) * (EMBED_TIME / 16)), dim3(32), 0, stream>>>(
        query, Wq, bq, qp, BB * LQ, EMBED_TIME, EMBED_TIME, 1.0f);

    // 2) K projection: [2048,128] @ [128,128] + bk
    gemm_wmma_f32<<<dim3((BB * LK / 16) * (EMBED_TIME / 16)), dim3(32), 0, stream>>>(
        key_ts, Wk, bk, kp, BB * LK, EMBED_TIME, EMBED_TIME, 1.0f);

    // 3) scores = qp_h @ kp_h^T / sqrt(dk)
    const float scale = 0.17677669529663687f;   // 1/sqrt(32)
    scores_wmma_f32<<<dim3((LQ / 16) * (LK / 16), BB * NUM_HEADS), dim3(32), 0, stream>>>(
        qp, kp, scores, scale);

    // 4) masked online softmax over Lk + weighted reduce -> x [4,256,256]
    mta_softmax_reduce<<<dim3(BB * LQ), dim3(256), 0, stream>>>(
        scores, value, mask, x);

    // 5) W3 row norms + stable descending permutation gather
    w3_rownorm<<<dim3(NHIDDEN), dim3(256), 0, stream>>>(W3, nrm);
    w3_permute<<<dim3(HD), dim3(256), 0, stream>>>(W3, nrm, W3p);

    // 6) out = x @ W3p + b3 : [1024,256] @ [256,256]
    gemm_wmma_f32<<<dim3((BB * LQ / 16) * (NHIDDEN / 16)), dim3(32), 0, stream>>>(
        x, W3p, b3, out, BB * LQ, NHIDDEN, HD, 1.0f);
}